// HierarchicalAttentionLayer_56160992362990
// MI455X (gfx1250) — compile-verified
//
#include <hip/hip_runtime.h>
#include <math.h>

#define B_ 64
#define L_ 512
#define D_ 768
#define LN_EPS 1e-6f

typedef __attribute__((ext_vector_type(2))) float v2f;
typedef __attribute__((ext_vector_type(4))) float v4f;
typedef __attribute__((ext_vector_type(8))) float v8f;

// ---------------------------------------------------------------------------
// Kernel 1: xt = emb_table[ids] @ W_dense + b_dense   (B x 2) * (2 x D)
// Mapped onto V_WMMA_F32_16X16X4_F32 (K padded 2 -> 4 with zeros).
// One wave (32 lanes) per 16x16 output tile. grid = (D/16, B/16).
// A (16x4 f32): lanes 0-15 hold K=0,1 in v[0],v[1]; lanes 16-31 hold K=2,3 (=0).
// B (4x16 f32): lanes 0-15 hold rows K=0,1 (N=lane); lanes 16-31 rows K=2,3 (=0).
// C/D (16x16 f32, 8 VGPRs): lane%16 = N, vgpr r = row M=r (+8 for hi lanes).
// ---------------------------------------------------------------------------
__global__ void proj_wmma_kernel(const int* __restrict__ ids,
                                 const float* __restrict__ emb_table,
                                 const float* __restrict__ W,      // (2, D)
                                 const float* __restrict__ bias,   // (D,)
                                 float* __restrict__ xt)           // (B, D)
{
    const int lane   = threadIdx.x;        // one wave per block
    const int m_base = blockIdx.y * 16;    // batch tile
    const int n_base = blockIdx.x * 16;    // output-column tile
    const int mi     = lane & 15;
    const bool lo    = lane < 16;

    // A-matrix fragment
    float e0 = 0.f, e1 = 0.f;
    if (lo) {
        const int id = ids[m_base + mi];
        e0 = emb_table[id * 2 + 0];
        e1 = emb_table[id * 2 + 1];
    }
    v2f a; a.x = e0; a.y = e1;             // hi lanes: K=2,3 are zero padding

    // B-matrix fragment
    const int n = n_base + mi;
    float b0 = 0.f, b1 = 0.f;
    if (lo) {
        b0 = W[0 * D_ + n];                // K = 0 row
        b1 = W[1 * D_ + n];                // K = 1 row
    }
    v2f bb; bb.x = b0; bb.y = b1;          // hi lanes: K=2,3 are zero padding

    v8f c = {};
    c = __builtin_amdgcn_wmma_f32_16x16x4_f32(
            /*neg_a=*/false, a, /*neg_b=*/false, bb,
            /*c_mod=*/(short)0, c, /*reuse_a=*/false, /*reuse_b=*/false);

    const float bd = bias[n];
#pragma unroll
    for (int r = 0; r < 8; ++r) {
        const int row = m_base + r + (lo ? 0 : 8);
        xt[row * D_ + n] = c[r] + bd;
    }
}

// ---------------------------------------------------------------------------
// Kernel 2: per-batch constants  xx[b] = sum(xt[b]^2), sx[b] = sum(xt[b])
// ---------------------------------------------------------------------------
__global__ void batch_stats_kernel(const float* __restrict__ xt,
                                   float* __restrict__ xx,
                                   float* __restrict__ sx)
{
    const int b = blockIdx.x;
    const int t = threadIdx.x;             // 256 threads, 8 waves
    const float* x = xt + b * D_;

    float s = 0.f, s2 = 0.f;
    for (int d = t; d < D_; d += 256) {
        const float v = x[d];
        s += v; s2 += v * v;
    }
#pragma unroll
    for (int off = 16; off; off >>= 1) {
        s  += __shfl_xor(s,  off, 32);
        s2 += __shfl_xor(s2, off, 32);
    }
    __shared__ float rs[8], rs2[8];
    if ((t & 31) == 0) { rs[t >> 5] = s; rs2[t >> 5] = s2; }
    __syncthreads();
    if (t == 0) {
        float S = 0.f, S2 = 0.f;
#pragma unroll
        for (int i = 0; i < 8; ++i) { S += rs[i]; S2 += rs2[i]; }
        sx[b] = S; xx[b] = S2;
    }
}

// ---------------------------------------------------------------------------
// Kernel 3: fused 2-token attention + layernorm, one block per (b,l) row.
// 192 threads (6 waves), one float4 (v4f) per thread covers D=768.
// pooler is read ONCE and out written ONCE -> non-temporal TH so the 192 MB
// stream does not evict the reused xt/gamma/beta from L2 (192 MB capacity).
// ---------------------------------------------------------------------------
__global__ void __launch_bounds__(192)
attn_ln_kernel(const float* __restrict__ pooler,
               const float* __restrict__ xt,
               const float* __restrict__ xx_arr,
               const float* __restrict__ sx_arr,
               const float* __restrict__ gamma,
               const float* __restrict__ beta,
               float* __restrict__ out)
{
    const int row = blockIdx.x;            // b*L + l
    const int b   = row >> 9;              // L = 512
    const int t   = threadIdx.x;           // 0..191

    // streaming (single-use) -> non-temporal load (TH=NT b128)
    const v4f p = __builtin_nontemporal_load(
        (const v4f*)(pooler + (size_t)row * D_) + t);
    // reused 512x per batch -> normal (RT) load, stays in L2
    const v4f x = ((const v4f*)(xt + (size_t)b * D_))[t];

    float pp = p.x * p.x + p.y * p.y + p.z * p.z + p.w * p.w;
    float px = p.x * x.x + p.y * x.y + p.z * x.z + p.w * x.w;
    float sp = p.x + p.y + p.z + p.w;

#pragma unroll
    for (int off = 16; off; off >>= 1) {
        pp += __shfl_xor(pp, off, 32);
        px += __shfl_xor(px, off, 32);
        sp += __shfl_xor(sp, off, 32);
    }

    __shared__ float red[3][6];
    __shared__ float sc[4];
    if ((t & 31) == 0) {
        const int w = t >> 5;
        red[0][w] = pp; red[1][w] = px; red[2][w] = sp;
    }
    __syncthreads();

    if (t == 0) {
        float PP = 0.f, PX = 0.f, SP = 0.f;
#pragma unroll
        for (int i = 0; i < 6; ++i) { PP += red[0][i]; PX += red[1][i]; SP += red[2][i]; }
        const float XX = xx_arr[b];
        const float SX = sx_arr[b];

        // softmax row 0 over [pp, px]; row 1 over [px, xx]
        const float m0  = fmaxf(PP, PX);
        const float e00 = __expf(PP - m0), e01 = __expf(PX - m0);
        const float a00 = e00 / (e00 + e01);
        const float m1  = fmaxf(PX, XX);
        const float e10 = __expf(PX - m1), e11 = __expf(XX - m1);
        const float a10 = e10 / (e10 + e11);

        const float wp = a00 + a10;        // weight on p in (out0 + out1)
        const float wx = 2.f - wp;         // weight on x

        const float mean = (wp * SP + wx * SX) * (1.f / D_);
        const float msq  = (wp * wp * PP + 2.f * wp * wx * PX + wx * wx * XX) * (1.f / D_);
        const float rstd = rsqrtf(msq - mean * mean + LN_EPS);
        sc[0] = wp; sc[1] = wx; sc[2] = mean; sc[3] = rstd;
    }
    __syncthreads();

    const float wp = sc[0], wx = sc[1], mean = sc[2], rstd = sc[3];
    const v4f g  = ((const v4f*)gamma)[t];
    const v4f be = ((const v4f*)beta)[t];

    v4f o;
    o.x = ((wp * p.x + wx * x.x) - mean) * rstd * g.x + be.x;
    o.y = ((wp * p.y + wx * x.y) - mean) * rstd * g.y + be.y;
    o.z = ((wp * p.z + wx * x.z) - mean) * rstd * g.z + be.z;
    o.w = ((wp * p.w + wx * x.w) - mean) * rstd * g.w + be.w;

    // streaming (single-use) -> non-temporal store (TH=NT b128)
    __builtin_nontemporal_store(o, (v4f*)(out + (size_t)row * D_) + t);
}

// ---------------------------------------------------------------------------
extern "C" void kernel_launch(void* const* d_in, const int* in_sizes, int n_in,
                              void* d_out, int out_size, void* d_ws, size_t ws_size,
                              hipStream_t stream)
{
    const int*   ids    = (const int*)  d_in[0];
    const float* pooler = (const float*)d_in[1];
    const float* emb    = (const float*)d_in[2];
    const float* Wd     = (const float*)d_in[3];
    const float* bd     = (const float*)d_in[4];
    const float* gamma  = (const float*)d_in[5];
    const float* beta   = (const float*)d_in[6];
    float* out = (float*)d_out;

    // workspace: xt (B*D floats) | xx (B) | sx (B)   ~197 KB
    float* xt = (float*)d_ws;
    float* xx = xt + B_ * D_;
    float* sx = xx + B_;

    proj_wmma_kernel<<<dim3(D_ / 16, B_ / 16), 32, 0, stream>>>(ids, emb, Wd, bd, xt);
    batch_stats_kernel<<<B_, 256, 0, stream>>>(xt, xx, sx);
    attn_ln_kernel<<<B_ * L_, 192, 0, stream>>>(pooler, xt, xx, sx, gamma, beta, out);
}